// EnzymesGNN_84026740179358
// MI455X (gfx1250) — compile-verified
//
#include <hip/hip_runtime.h>
#include <math.h>

// ---------------- constants from the reference ----------------
#define FD    64      // node feature dim D
#define FD2   128     // 2*D
#define HIDN  128     // cls hidden
#define NCLSN 6

#define SELU_SCALE 1.0507009873554805f
#define SELU_SA    1.7580993408473766f   // scale*alpha

typedef __attribute__((ext_vector_type(2))) float v2f;
typedef __attribute__((ext_vector_type(8))) float v8f;

// w[k] = sum_i a_vec[i] * W_att[i][k], k in [0,128)
__global__ void k_wred(const float* __restrict__ W_att, const float* __restrict__ a_vec,
                       float* __restrict__ w) {
    int k = threadIdx.x;             // 128 threads
    float s = 0.f;
    #pragma unroll 8
    for (int i = 0; i < 64; ++i) s += a_vec[i] * W_att[i * FD2 + k];
    w[k] = s;
}

__global__ void k_zero_u32(unsigned* __restrict__ p, int n) {
    int i = blockIdx.x * blockDim.x + threadIdx.x;
    if (i < n) p[i] = 0u;
}

__global__ void k_count(const int* __restrict__ col, unsigned* __restrict__ cnt, int E) {
    int i = blockIdx.x * blockDim.x + threadIdx.x;
    if (i < E) atomicAdd(&cnt[col[i]], 1u);     // integer atomics: deterministic
}

__global__ void k_dsqrt(const unsigned* __restrict__ cnt, float* __restrict__ ds, int n) {
    int i = blockIdx.x * blockDim.x + threadIdx.x;
    if (i < n) ds[i] = sqrtf((float)cnt[i]);
}

// per-node attention scalars: ep = exp(x.w_a), eq = exp(x.w_b)
__global__ void k_pq(const float* __restrict__ x, const float* __restrict__ w,
                     float* __restrict__ ep, float* __restrict__ eq, int N) {
    int n = blockIdx.x * blockDim.x + threadIdx.x;
    if (n >= N) return;
    const float* xr = x + (size_t)n * FD;
    float p = 0.f, q = 0.f;
    #pragma unroll 8
    for (int k = 0; k < FD; ++k) {
        float v = xr[k];
        p += w[k] * v;
        q += w[FD + k] * v;
    }
    ep[n] = expf(p);
    eq[n] = expf(q);
}

// S[n] = sum over incoming edges (contiguous block of DEG) of ep[row[e]]
__global__ void k_S(const int* __restrict__ row, const float* __restrict__ ep,
                    float* __restrict__ S, int N, int DEG) {
    int n = blockIdx.x * blockDim.x + threadIdx.x;
    if (n >= N) return;
    const int* r = row + (size_t)n * DEG;
    float s = 0.f;
    for (int e = 0; e < DEG; ++e) s += ep[r[e]];
    S[n] = s;
}

// wave-per-node aggregation:
// agg[n,:] = sum_e coef_e * x[src_e,:],  coef_e = ep[src]*eq[n]/(eq[src]*S[src]*ds[src]*ds[n])
// bsum[n]  = sum_e 1/(ds[src]*ds[n])
__global__ void k_agg(const float* __restrict__ x, const int* __restrict__ row,
                      const float* __restrict__ ep, const float* __restrict__ eq,
                      const float* __restrict__ S, const float* __restrict__ ds,
                      float* __restrict__ agg, float* __restrict__ bsum,
                      int N, int DEG) {
    int wave = (blockIdx.x * blockDim.x + threadIdx.x) >> 5;
    int lane = threadIdx.x & 31;
    if (wave >= N) return;
    const int n = wave;
    const float dn  = ds[n];
    const float eqn = eq[n];
    const int* r = row + (size_t)n * DEG;
    float ax = 0.f, ay = 0.f, bs = 0.f;
    for (int e = 0; e < DEG; ++e) {
        int src = r[e];                                   // lane-uniform
        float inv_ns = 1.f / (ds[src] * dn);
        float coef = (ep[src] * eqn) / (eq[src] * S[src]) * inv_ns;
        const v2f xv = *(const v2f*)(x + (size_t)src * FD + lane * 2);  // 8B gather
        ax += coef * xv.x;
        ay += coef * xv.y;
        bs += inv_ns;
    }
    v2f* o = (v2f*)(agg + (size_t)n * FD + lane * 2);
    v2f ov; ov.x = ax; ov.y = ay;
    *o = ov;
    if (lane == 0) bsum[n] = bs;
}

// h = selu(agg @ W_lin.T + bsum (x) b_lin)   via V_WMMA_F32_16X16X4_F32
// one wave per 16x16 output tile; block = 4 waves = 4 column tiles of one row tile.
// A(16x4) lane layout:  lane L holds A[L%16][ (L/16)*2 + v ], v=0,1
// B(4x16) lane layout:  lane L holds B[(L/16)*2 + v][L%16]; B[k][j] = W_lin[j][k]
// C/D(16x16):           lane L, VGPR v holds C[(L/16)*8 + v][L%16]
__global__ void k_gemm_selu(const float* __restrict__ agg, const float* __restrict__ Wl,
                            const float* __restrict__ bl,  const float* __restrict__ bsum,
                            float* __restrict__ h) {
    const int lane = threadIdx.x & 31;
    const int wv   = threadIdx.x >> 5;       // column tile 0..3
    const int m0   = blockIdx.x * 16;
    const int j0   = wv * 16;
    const int half = lane >> 4;
    const int lm   = lane & 15;

    v8f c;
    const float blj = bl[j0 + lm];
    #pragma unroll
    for (int v = 0; v < 8; ++v) c[v] = bsum[m0 + half * 8 + v] * blj;

    const float* arow = agg + (size_t)(m0 + lm) * FD + half * 2;
    const float* brow = Wl  + (size_t)(j0 + lm) * FD + half * 2;

    #pragma unroll
    for (int k0 = 0; k0 < FD; k0 += 4) {
        v2f a, b;
        a.x = arow[k0]; a.y = arow[k0 + 1];
        b.x = brow[k0]; b.y = brow[k0 + 1];
        // (neg_a, A, neg_b, B, c_mod, C, reuse_a, reuse_b)
        c = __builtin_amdgcn_wmma_f32_16x16x4_f32(false, a, false, b, (short)0, c, false, false);
    }

    #pragma unroll
    for (int v = 0; v < 8; ++v) {
        float o = c[v];
        o = (o > 0.f) ? SELU_SCALE * o : SELU_SA * (expf(o) - 1.f);
        h[(size_t)(m0 + half * 8 + v) * FD + j0 + lm] = o;
    }
}

// mean pool per graph (batch = contiguous blocks of NPG nodes)
__global__ void k_pool(const float* __restrict__ h, float* __restrict__ pooled, int NPG) {
    int g = blockIdx.x;
    int d = threadIdx.x;                     // 64 threads
    const float* base = h + (size_t)g * NPG * FD + d;
    float s = 0.f;
    for (int m = 0; m < NPG; ++m) s += base[(size_t)m * FD];
    pooled[g * FD + d] = s / (float)NPG;
}

// classifier: selu(pooled@W1.T+b1) @ W2.T + b2 -> softmax
__global__ void k_cls(const float* __restrict__ pooled,
                      const float* __restrict__ W1, const float* __restrict__ b1,
                      const float* __restrict__ W2, const float* __restrict__ b2,
                      float* __restrict__ out) {
    __shared__ float sp[FD];
    __shared__ float sh[HIDN];
    __shared__ float sl[NCLSN];
    const int g = blockIdx.x;
    const int t = threadIdx.x;               // 128 threads
    if (t < FD) sp[t] = pooled[g * FD + t];
    __syncthreads();
    {
        float a = b1[t];
        const float* wr = W1 + (size_t)t * FD;
        #pragma unroll 8
        for (int k = 0; k < FD; ++k) a += wr[k] * sp[k];
        sh[t] = (a > 0.f) ? SELU_SCALE * a : SELU_SA * (expf(a) - 1.f);
    }
    __syncthreads();
    if (t < NCLSN) {
        float l = b2[t];
        const float* wr = W2 + (size_t)t * HIDN;
        #pragma unroll 8
        for (int k = 0; k < HIDN; ++k) l += wr[k] * sh[k];
        sl[t] = l;
    }
    __syncthreads();
    if (t == 0) {
        float mx = sl[0];
        for (int c2 = 1; c2 < NCLSN; ++c2) mx = fmaxf(mx, sl[c2]);
        float e[NCLSN], s = 0.f;
        for (int c2 = 0; c2 < NCLSN; ++c2) { e[c2] = expf(sl[c2] - mx); s += e[c2]; }
        float inv = 1.f / s;
        for (int c2 = 0; c2 < NCLSN; ++c2) out[g * NCLSN + c2] = e[c2] * inv;
    }
}

extern "C" void kernel_launch(void* const* d_in, const int* in_sizes, int n_in,
                              void* d_out, int out_size, void* d_ws, size_t ws_size,
                              hipStream_t stream) {
    const float* x     = (const float*)d_in[0];
    const float* W_lin = (const float*)d_in[1];
    const float* b_lin = (const float*)d_in[2];
    const float* W_att = (const float*)d_in[3];
    const float* a_vec = (const float*)d_in[4];
    const float* W1    = (const float*)d_in[5];
    const float* b1    = (const float*)d_in[6];
    const float* W2    = (const float*)d_in[7];
    const float* b2    = (const float*)d_in[8];
    const int*   ei    = (const int*)d_in[9];
    float* out = (float*)d_out;

    const int N   = in_sizes[0] / FD;        // 40000
    const int E   = in_sizes[9] / 2;         // 1,280,000
    const int DEG = E / N;                   // 32
    const int G   = out_size / NCLSN;        // 200
    const int NPG = N / G;                   // 200

    const int* row = ei;
    const int* col = ei + E;

    // -------- workspace carve (256B aligned regions) --------
    char* base = (char*)d_ws;
    size_t off = 0;
    auto carve = [&](size_t bytes) -> char* {
        char* p = base + off;
        off += (bytes + 255) & ~(size_t)255;
        return p;
    };
    float*    w_red  = (float*)   carve(FD2 * sizeof(float));
    float*    ep     = (float*)   carve((size_t)N * sizeof(float));
    float*    eq     = (float*)   carve((size_t)N * sizeof(float));
    float*    Sb     = (float*)   carve((size_t)N * sizeof(float));
    float*    dsq    = (float*)   carve((size_t)N * sizeof(float));
    float*    bsum   = (float*)   carve((size_t)N * sizeof(float));
    unsigned* cnt    = (unsigned*)carve((size_t)N * sizeof(unsigned));
    float*    agg    = (float*)   carve((size_t)N * FD * sizeof(float));
    float*    hbuf   = (float*)   carve((size_t)N * FD * sizeof(float));
    float*    pooled = (float*)   carve((size_t)G * FD * sizeof(float));
    (void)ws_size; (void)n_in;

    // 1) w = W_att.T @ a_vec
    k_wred<<<1, FD2, 0, stream>>>(W_att, a_vec, w_red);

    // 2) degrees (integer atomics -> deterministic), then sqrt
    k_zero_u32<<<(N + 255) / 256, 256, 0, stream>>>(cnt, N);
    k_count<<<(E + 255) / 256, 256, 0, stream>>>(col, cnt, E);
    k_dsqrt<<<(N + 255) / 256, 256, 0, stream>>>(cnt, dsq, N);

    // 3) per-node attention scalars
    k_pq<<<(N + 127) / 128, 128, 0, stream>>>(x, w_red, ep, eq, N);

    // 4) softmax denominators per target node
    k_S<<<(N + 127) / 128, 128, 0, stream>>>(row, ep, Sb, N, DEG);

    // 5) weighted gather-aggregate (wave per node)
    k_agg<<<(N * 32 + 255) / 256, 256, 0, stream>>>(x, row, ep, eq, Sb, dsq, agg, bsum, N, DEG);

    // 6) dense GEMM + bias + SELU via f32 WMMA (16 wmma per 16x16 tile)
    k_gemm_selu<<<N / 16, 128, 0, stream>>>(agg, W_lin, b_lin, bsum, hbuf);

    // 7) global mean pool (contiguous graphs)
    k_pool<<<G, FD, 0, stream>>>(hbuf, pooled, NPG);

    // 8) classifier MLP + softmax
    k_cls<<<G, HIDN, 0, stream>>>(pooled, W1, b1, W2, b2, out);
}